// VQVAE_Pose_ML_63213328663043
// MI455X (gfx1250) — compile-verified
//
#include <hip/hip_runtime.h>
#include <hip/hip_bf16.h>

typedef float v2f __attribute__((ext_vector_type(2)));
typedef float v8f __attribute__((ext_vector_type(8)));

static constexpr int NROWS = 16 * 4096;   // B*T = 65536
static constexpr int ED_ = 64;
static constexpr int K_  = 512;
static constexpr int NL_ = 4;

__device__ __forceinline__ v8f wmma_f32x4(v2f a, v2f b, v8f c) {
  // D = A(16x4 f32) * B(4x16 f32) + C(16x16 f32)
  return __builtin_amdgcn_wmma_f32_16x16x4_f32(false, a, false, b, (short)0, c, false, false);
}

// One wave32 computes a 16x16 tile of Y = act_out( act_in(X) @ W^T + bias [+ R] ).
// X:[rows,Cin] row-major, W:[Cout,Cin] row-major. 8 waves / block (256 thr), 128 rows / block.
template<bool RELU_IN, bool RELU_OUT, bool HAS_RES, bool KVEC>
__global__ __launch_bounds__(256) void gemm16(
    const float* __restrict__ X, const float* __restrict__ W,
    const float* __restrict__ bias, const float* __restrict__ R,
    float* __restrict__ Y, int Cin, int Cout) {
  const int lane = threadIdx.x & 31;
  const int wave = threadIdx.x >> 5;
  const int r  = lane & 15;        // row-within-tile for A, col row for B, col for D
  const int hk = lane >> 4;        // lane half selects K sub-pair / D row group
  const int colbase = blockIdx.x << 4;
  const int rowbase = ((blockIdx.y << 3) + wave) << 4;

  const float* __restrict__ xptr = X + (size_t)(rowbase + r) * Cin;
  const int wrow  = colbase + r;
  const int wrowc = (wrow < Cout) ? wrow : (Cout - 1);   // clamp: garbage B cols never stored
  const float* __restrict__ wptr = W + (size_t)wrowc * Cin;

  v8f acc = {0.f,0.f,0.f,0.f,0.f,0.f,0.f,0.f};

  for (int k0 = 0; k0 < Cin; k0 += 4) {
    float x0,x1,x2,x3,w0,w1,w2,w3;
    if (KVEC) {  // Cin % 4 == 0: aligned float4 loads
      const float4 xv = *reinterpret_cast<const float4*>(xptr + k0);
      const float4 wv = *reinterpret_cast<const float4*>(wptr + k0);
      x0=xv.x; x1=xv.y; x2=xv.z; x3=xv.w;
      w0=wv.x; w1=wv.y; w2=wv.z; w3=wv.w;
    } else {     // ragged Cin (165): guarded scalar loads, zero-fill tail
      x0 = (k0+0 < Cin) ? xptr[k0+0] : 0.f;
      x1 = (k0+1 < Cin) ? xptr[k0+1] : 0.f;
      x2 = (k0+2 < Cin) ? xptr[k0+2] : 0.f;
      x3 = (k0+3 < Cin) ? xptr[k0+3] : 0.f;
      w0 = (k0+0 < Cin) ? wptr[k0+0] : 0.f;
      w1 = (k0+1 < Cin) ? wptr[k0+1] : 0.f;
      w2 = (k0+2 < Cin) ? wptr[k0+2] : 0.f;
      w3 = (k0+3 < Cin) ? wptr[k0+3] : 0.f;
    }
    v2f a, b;
    a.x = hk ? x2 : x0;  a.y = hk ? x3 : x1;   // A[m][k0+2*hk+j]
    b.x = hk ? w2 : w0;  b.y = hk ? w3 : w1;   // B[k0+2*hk+j][n] = W[n][k]
    if (RELU_IN) { a.x = fmaxf(a.x, 0.f); a.y = fmaxf(a.y, 0.f); }
    acc = wmma_f32x4(a, b, acc);
  }

  const int n = colbase + r;                   // D: lane -> col n, vgpr v -> row v+8*hk
  if (n < Cout) {
    const float bv = bias[n];
#pragma unroll
    for (int v = 0; v < 8; ++v) {
      const int m = rowbase + v + (hk << 3);
      float val = acc[v] + bv;
      if (HAS_RES) val += R[(size_t)m * Cout + n];
      if (RELU_OUT) val = fmaxf(val, 0.f);
      Y[(size_t)m * Cout + n] = val;           // lanes 0-15 / 16-31 each write 64B runs
    }
  }
}

// One wave handles 16 rows of residual VQ: scores = ||e||^2 - 2*res.e via WMMA,
// argmin over 512 codes, gather q, emit q/new residual, accumulate diff.
__global__ __launch_bounds__(256) void vq_level(
    const float* __restrict__ resin,
    const float* __restrict__ embed,    // [64][512] (level slice of embeds)
    const float* __restrict__ embedT,   // [512][64] transposed codebook
    const float* __restrict__ eksq,     // [512] code norms
    float* __restrict__ q_out,          // may be null
    float* __restrict__ resout,
    float* __restrict__ diff_acc) {
  const int lane = threadIdx.x & 31;
  const int wave = threadIdx.x >> 5;
  const int r  = lane & 15;
  const int hk = lane >> 4;
  const int rowbase = ((blockIdx.x << 3) + wave) << 4;

  // A fragments: full 64-dim rows resident in VGPRs; also tile's sum of res^2
  v2f afrag[16];
  float rsq = 0.f;
  const float* __restrict__ xptr = resin + (size_t)(rowbase + r) * ED_;
#pragma unroll
  for (int s = 0; s < 16; ++s) {
    const float4 xv = *reinterpret_cast<const float4*>(xptr + 4 * s);
    v2f a; a.x = hk ? xv.z : xv.x; a.y = hk ? xv.w : xv.y;
    afrag[s] = a;
    rsq += a.x * a.x + a.y * a.y;
  }

  float minval[8]; int minidx[8];
#pragma unroll
  for (int v = 0; v < 8; ++v) { minval[v] = 3.4e38f; minidx[v] = 0; }

  for (int t = 0; t < 32; ++t) {               // 32 tiles of 16 codes
    const int code = (t << 4) + r;
    const float e2 = eksq[code];
    v8f acc = {0.f,0.f,0.f,0.f,0.f,0.f,0.f,0.f};
#pragma unroll
    for (int s = 0; s < 16; ++s) {             // K = 64
      v2f b;
      b.x = embed[(size_t)((s << 2) + (hk << 1) + 0) * K_ + code];  // coalesced over lanes
      b.y = embed[(size_t)((s << 2) + (hk << 1) + 1) * K_ + code];
      acc = wmma_f32x4(afrag[s], b, acc);
    }
#pragma unroll
    for (int v = 0; v < 8; ++v) {
      const float score = fmaf(-2.f, acc[v], e2);   // ||e||^2 - 2*res.e
      if (score < minval[v]) { minval[v] = score; minidx[v] = code; }  // strict: keeps lowest code
    }
  }

  // argmin butterfly within each 16-lane half (xor 1,2,4,8 stay in-half); tie -> lower index
#pragma unroll
  for (int m1 = 1; m1 <= 8; m1 <<= 1) {
#pragma unroll
    for (int v = 0; v < 8; ++v) {
      const float ov = __shfl_xor(minval[v], m1, 32);
      const int   oi = __shfl_xor(minidx[v], m1, 32);
      if (ov < minval[v] || (ov == minval[v] && oi < minidx[v])) { minval[v] = ov; minidx[v] = oi; }
    }
  }

  // diff: sum_rows(min score) + sum(res^2) == sum_rows ||q - res||^2
  float smin = 0.f;
#pragma unroll
  for (int v = 0; v < 8; ++v) smin += minval[v];
  smin += __shfl_xor(smin, 16, 32);
#pragma unroll
  for (int m1 = 1; m1 <= 16; m1 <<= 1) rsq += __shfl_xor(rsq, m1, 32);
  if (lane == 0) atomicAdd(diff_acc, smin + rsq);

  // broadcast per-row winners (half0 holds rows 0-7, half1 rows 8-15)
  int idxs[16];
#pragma unroll
  for (int v = 0; v < 8; ++v) {
    idxs[v]     = __shfl(minidx[v], 0, 32);
    idxs[v + 8] = __shfl(minidx[v], 16, 32);
  }

  // gather q (coalesced float2 via transposed codebook), straight-through numerics
  const int d = lane << 1;
#pragma unroll
  for (int m = 0; m < 16; ++m) {
    const size_t off = (size_t)(rowbase + m) * ED_ + d;
    const float2 rv = *reinterpret_cast<const float2*>(resin + off);
    const float2 ev = *reinterpret_cast<const float2*>(embedT + (size_t)idxs[m] * ED_ + d);
    const float q0 = rv.x + (ev.x - rv.x);     // q_st = res + (q - res)
    const float q1 = rv.y + (ev.y - rv.y);
    float2 nr; nr.x = rv.x - q0; nr.y = rv.y - q1;
    *reinterpret_cast<float2*>(resout + off) = nr;
    if (q_out) { float2 qv; qv.x = q0; qv.y = q1; *reinterpret_cast<float2*>(q_out + off) = qv; }
  }
}

__global__ void vq_prep(const float* __restrict__ embeds, float* __restrict__ embedT,
                        float* __restrict__ eksq, float* __restrict__ diff_acc) {
  const int k = blockIdx.x * blockDim.x + threadIdx.x;
  if (k < NL_) diff_acc[k] = 0.f;
  if (k >= NL_ * K_) return;
  const int lvl = k >> 9, code = k & (K_ - 1);
  const float* e = embeds + (size_t)lvl * ED_ * K_;
  float s = 0.f;
  for (int dd = 0; dd < ED_; ++dd) {
    const float v = e[(size_t)dd * K_ + code];
    embedT[((size_t)lvl * K_ + code) * ED_ + dd] = v;
    s = fmaf(v, v, s);
  }
  eksq[k] = s;
}

__global__ void vq_finalize(const float* __restrict__ diff_acc, float* __restrict__ out) {
  if (threadIdx.x == 0 && blockIdx.x == 0) {
    float s = 0.f;
    for (int i = 0; i < NL_; ++i) s += diff_acc[i];
    out[0] = s * (0.25f / (float)(NROWS * ED_));   // mean over levels of per-level means
  }
}

extern "C" void kernel_launch(void* const* d_in, const int* in_sizes, int n_in,
                              void* d_out, int out_size, void* d_ws, size_t ws_size,
                              hipStream_t stream) {
  (void)in_sizes; (void)n_in; (void)out_size; (void)ws_size;
  const float* x          = (const float*)d_in[0];
  const float* enc_w1     = (const float*)d_in[1];
  const float* enc_b1     = (const float*)d_in[2];
  const float* enc_w2     = (const float*)d_in[3];
  const float* enc_b2     = (const float*)d_in[4];
  const float* enc_w3     = (const float*)d_in[5];
  const float* enc_b3     = (const float*)d_in[6];
  const float* enc_res_wa = (const float*)d_in[7];
  const float* enc_res_ba = (const float*)d_in[8];
  const float* enc_res_wb = (const float*)d_in[9];
  const float* enc_res_bb = (const float*)d_in[10];
  const float* q_w        = (const float*)d_in[11];
  const float* q_b        = (const float*)d_in[12];
  const float* embeds     = (const float*)d_in[13];
  const float* dec_w1     = (const float*)d_in[14];
  const float* dec_b1     = (const float*)d_in[15];
  const float* dec_res_wa = (const float*)d_in[16];
  const float* dec_res_ba = (const float*)d_in[17];
  const float* dec_res_wb = (const float*)d_in[18];
  const float* dec_res_bb = (const float*)d_in[19];
  const float* dec_w2     = (const float*)d_in[20];
  const float* dec_b2     = (const float*)d_in[21];
  const float* dec_w3     = (const float*)d_in[22];
  const float* dec_b3     = (const float*)d_in[23];

  const int N = NROWS;
  float* ws    = (float*)d_ws;
  float* bufA  = ws;                               // [N,256]
  float* bufB  = bufA  + (size_t)256 * N;          // [N,256]
  float* bufT  = bufB  + (size_t)256 * N;          // [N,32]
  float* qbuf  = bufT  + (size_t)32  * N;          // [N,64]
  float* rA    = qbuf  + (size_t)64  * N;          // [N,64]
  float* rB    = rA    + (size_t)64  * N;          // [N,64]
  float* mainq = rB    + (size_t)64  * N;          // [N,64]
  float* embT  = mainq + (size_t)64  * N;          // [4,512,64]
  float* eksq  = embT  + (size_t)NL_ * K_ * ED_;   // [4,512]
  float* dacc  = eksq  + (size_t)NL_ * K_;         // [4]

  const dim3 blk(256, 1, 1);
  const unsigned GY = N / 128;                     // 512: 8 waves * 16 rows per block

  vq_prep<<<dim3((NL_*K_ + 255)/256, 1, 1), blk, 0, stream>>>(embeds, embT, eksq, dacc);

  // ---- encoder ----
  gemm16<false,true ,false,false><<<dim3( 8,GY,1), blk, 0, stream>>>(x,    enc_w1, enc_b1, nullptr, bufA, 165, 128);
  gemm16<false,true ,false,true ><<<dim3(16,GY,1), blk, 0, stream>>>(bufA, enc_w2, enc_b2, nullptr, bufB, 128, 256);
  gemm16<false,false,false,true ><<<dim3(16,GY,1), blk, 0, stream>>>(bufB, enc_w3, enc_b3, nullptr, bufA, 256, 256);
  gemm16<true ,true ,false,true ><<<dim3( 2,GY,1), blk, 0, stream>>>(bufA, enc_res_wa,          enc_res_ba,      nullptr, bufT, 256, 32);
  gemm16<false,false,true ,true ><<<dim3(16,GY,1), blk, 0, stream>>>(bufT, enc_res_wb,          enc_res_bb,      bufA,    bufB, 32, 256);
  gemm16<true ,true ,false,true ><<<dim3( 2,GY,1), blk, 0, stream>>>(bufB, enc_res_wa + 32*256, enc_res_ba + 32, nullptr, bufT, 256, 32);
  gemm16<false,false,true ,true ><<<dim3(16,GY,1), blk, 0, stream>>>(bufT, enc_res_wb + 256*32, enc_res_bb + 256, bufB,   bufA, 32, 256);
  gemm16<true ,false,false,true ><<<dim3( 4,GY,1), blk, 0, stream>>>(bufA, q_w, q_b, nullptr, qbuf, 256, 64);

  // ---- residual VQ (4 levels) ----
  vq_level<<<dim3(GY,1,1), blk, 0, stream>>>(qbuf, embeds + 0*ED_*K_, embT + 0*K_*ED_, eksq + 0*K_, mainq,   rA, dacc + 0);
  vq_level<<<dim3(GY,1,1), blk, 0, stream>>>(rA,   embeds + 1*ED_*K_, embT + 1*K_*ED_, eksq + 1*K_, nullptr, rB, dacc + 1);
  vq_level<<<dim3(GY,1,1), blk, 0, stream>>>(rB,   embeds + 2*ED_*K_, embT + 2*K_*ED_, eksq + 2*K_, nullptr, rA, dacc + 2);
  vq_level<<<dim3(GY,1,1), blk, 0, stream>>>(rA,   embeds + 3*ED_*K_, embT + 3*K_*ED_, eksq + 3*K_, nullptr, rB, dacc + 3);

  // ---- decoder ----
  float* dout = (float*)d_out;
  gemm16<false,false,false,true ><<<dim3(16,GY,1), blk, 0, stream>>>(mainq, dec_w1, dec_b1, nullptr, bufA, 64, 256);
  gemm16<true ,true ,false,true ><<<dim3( 2,GY,1), blk, 0, stream>>>(bufA, dec_res_wa,          dec_res_ba,      nullptr, bufT, 256, 32);
  gemm16<false,false,true ,true ><<<dim3(16,GY,1), blk, 0, stream>>>(bufT, dec_res_wb,          dec_res_bb,      bufA,    bufB, 32, 256);
  gemm16<true ,true ,false,true ><<<dim3( 2,GY,1), blk, 0, stream>>>(bufB, dec_res_wa + 32*256, dec_res_ba + 32, nullptr, bufT, 256, 32);
  gemm16<false,false,true ,true ><<<dim3(16,GY,1), blk, 0, stream>>>(bufT, dec_res_wb + 256*32, dec_res_bb + 256, bufB,   bufA, 32, 256);
  gemm16<true ,true ,false,true ><<<dim3( 8,GY,1), blk, 0, stream>>>(bufA, dec_w2, dec_b2, nullptr, bufB, 256, 128);
  gemm16<false,false,false,true ><<<dim3(11,GY,1), blk, 0, stream>>>(bufB, dec_w3, dec_b3, nullptr, dout, 128, 165);

  vq_finalize<<<dim3(1,1,1), dim3(1,1,1), 0, stream>>>(dacc, dout + (size_t)N * 165);
}